// make_blocks_32521492365666
// MI455X (gfx1250) — compile-verified
//
#include <hip/hip_runtime.h>
#include <cstdint>

// Problem constants (from reference): B,P,S=32, C=64, R=2048, L=1024
#define NB 32
#define NP 32
#define NS 32
#define NC 64
#define NR 2048
#define NL 1024

#define ROW_SPAN 129u            // 2C+1 floats per (i,j)
#define I_SPAN   4128u           // S * 129 floats per i
#define BLK_F4   33024u          // S*S*129/4 float4 per (b,p)

typedef float f4 __attribute__((ext_vector_type(4)));
typedef int   v4i __attribute__((vector_size(16)));   // matches builtin param type

#if defined(__gfx1250__) && __has_builtin(__builtin_amdgcn_global_load_async_to_lds_b128)
#define USE_ASYNC_LDS 1
#else
#define USE_ASYNC_LDS 0
#endif

typedef __attribute__((address_space(1))) v4i gv4i;   // prints as "__device__" in HIP diags
typedef __attribute__((address_space(3))) v4i lv4i;   // prints as "__shared__"

__device__ static inline gv4i* to_global(const void* p) {
    return (gv4i*)(unsigned long long)(uintptr_t)p;
}
__device__ static inline lv4i* to_lds(void* p) {
    // generic LDS address keeps LDS byte offset in low 32 bits (CDNA5 aperture rule)
    return (lv4i*)(unsigned)(uintptr_t)p;
}

__global__ __launch_bounds__(256)
void make_blocks_32521492365666_kernel(const float* __restrict__ seq1M,
                                       const float* __restrict__ seq2M,
                                       const int*   __restrict__ patches,
                                       const float* __restrict__ geo,
                                       float*       __restrict__ out) {
    __shared__ float lrows[NS * NC];   // 2048 f32 = 8 KB
    __shared__ float lcols[NS * NC];   // 2048 f32 = 8 KB
    __shared__ float lgeo [NS * NS];   // 1024 f32 = 4 KB

    const unsigned bp  = blockIdx.x;        // 0..1023  == b*P + p
    const unsigned b   = bp >> 5;           // /32
    const unsigned tid = threadIdx.x;

    const int r0 = patches[bp * 2 + 0];
    const int c0 = patches[bp * 2 + 1];

    const float* rsrc = seq1M + ((size_t)b * NR + (unsigned)r0) * NC; // 2048 contiguous f32
    const float* csrc = seq2M + ((size_t)b * NL + (unsigned)c0) * NC; // 2048 contiguous f32
    const float* gsrc = geo   + (size_t)bp * (NS * NS);               // 1024 contiguous f32

#if USE_ASYNC_LDS
    {
        const unsigned o = tid * 4u;  // float offset of this thread's 16B chunk
        // rows: 512 x b128 -> 2 per thread
        __builtin_amdgcn_global_load_async_to_lds_b128(to_global(rsrc + o),
                                                       to_lds(&lrows[o]), 0, 0);
        __builtin_amdgcn_global_load_async_to_lds_b128(to_global(rsrc + 1024u + o),
                                                       to_lds(&lrows[1024u + o]), 0, 0);
        // cols: 512 x b128 -> 2 per thread
        __builtin_amdgcn_global_load_async_to_lds_b128(to_global(csrc + o),
                                                       to_lds(&lcols[o]), 0, 0);
        __builtin_amdgcn_global_load_async_to_lds_b128(to_global(csrc + 1024u + o),
                                                       to_lds(&lcols[1024u + o]), 0, 0);
        // geo: 256 x b128 -> 1 per thread
        __builtin_amdgcn_global_load_async_to_lds_b128(to_global(gsrc + o),
                                                       to_lds(&lgeo[o]), 0, 0);
    }
#if __has_builtin(__builtin_amdgcn_s_wait_asynccnt)
    __builtin_amdgcn_s_wait_asynccnt(0);
#else
    asm volatile("s_wait_asynccnt 0x0" ::: "memory");
#endif
#else
    for (unsigned q = tid; q < NS * NC; q += 256u) lrows[q] = rsrc[q];
    for (unsigned q = tid; q < NS * NC; q += 256u) lcols[q] = csrc[q];
    for (unsigned q = tid; q < NS * NS; q += 256u) lgeo[q]  = gsrc[q];
#endif
    __syncthreads();

    // ---- Stream 516 KB of output per block as coalesced NT float4 stores ----
    // out[b,p,i,j,k]: k<64 -> lrows[j*64+k]; 64<=k<128 -> lcols[i*64+k-64]; k==128 -> lgeo[i*32+j]
    f4* out4 = reinterpret_cast<f4*>(out) + (size_t)bp * BLK_F4;

    for (unsigned q = tid; q < BLK_F4; q += 256u) {
        const unsigned t   = q * 4u;              // flat float index in this block
        const unsigned i   = t / I_SPAN;          // 4128 | t never straddles i (4128 % 4 == 0)
        const unsigned rem = t - i * I_SPAN;
        unsigned jj = rem / ROW_SPAN;
        unsigned kk = rem - jj * ROW_SPAN;

        float tmp[4];
#pragma unroll
        for (int d = 0; d < 4; ++d) {
            float x;
            if (kk < 64u)       x = lrows[jj * 64u + kk];
            else if (kk < 128u) x = lcols[i  * 64u + (kk - 64u)];
            else                x = lgeo [i  * 32u + jj];
            tmp[d] = x;
            if (++kk == ROW_SPAN) { kk = 0u; ++jj; }   // jj stays < 32 within a vector
        }
        f4 v;
        v.x = tmp[0]; v.y = tmp[1]; v.z = tmp[2]; v.w = tmp[3];
        __builtin_nontemporal_store(v, out4 + q);      // NT: 541 MB write-once stream
    }
}

extern "C" void kernel_launch(void* const* d_in, const int* in_sizes, int n_in,
                              void* d_out, int out_size, void* d_ws, size_t ws_size,
                              hipStream_t stream) {
    const float* seq1M   = (const float*)d_in[0];
    const float* seq2M   = (const float*)d_in[1];
    const int*   patches = (const int*)  d_in[2];
    const float* geo     = (const float*)d_in[3];
    float*       out     = (float*)d_out;

    dim3 grid(NB * NP);   // 1024 blocks: one per (b,p)
    dim3 block(256);      // 8 wave32
    hipLaunchKernelGGL(make_blocks_32521492365666_kernel, grid, block, 0, stream,
                       seq1M, seq2M, patches, geo, out);
}